// AFNO2D_49495203119184
// MI455X (gfx1250) — compile-verified
//
#include <hip/hip_runtime.h>
#include <math.h>

// ---------------------------------------------------------------------------
// AFNO2D on MI455X (gfx1250): truncated-DFT-as-WMMA pipeline.
//   stage1: rDFT_w   (per b,h)  [32x128]x[128x768]        -> S1 (b,h,kw,c) r/i
//   stage2: DFT_h    (per b)    [32x128]x[128x24576]      -> S2 (b,kh,kw,c) r/i
//           (B operand streamed via async global->LDS double buffering)
//   stage3: block-diag complex MLP (LDS-staged o1, GELU)  -> S3 (b,kh,kw,c) r/i
//   stage4: iDFT_h   (per b)    [128x32]x[32x24576]       -> S4 (b,h,kw,c) r/i
//   stage5: irDFT_w + residual  [128x32]x[32x768] + x     -> out (b,h,w,c)
// All GEMMs use V_WMMA_F32_16X16X4_F32 (fp32 end to end).
// ---------------------------------------------------------------------------

typedef float v2f __attribute__((ext_vector_type(2)));
typedef float v8f __attribute__((ext_vector_type(8)));
typedef int   v4i __attribute__((ext_vector_type(4)));

#define NB_BLK 8
#define BS_    96
#define C_     768
#define KM_    32          // kept modes per axis
#define N2_    (KM_ * C_)  // 24576, stage2/4 GEMM width

// twiddle-table offsets (in floats) inside workspace
#define T_FW_R 0
#define T_FW_I 4096
#define T_FH_R 8192
#define T_FH_I 12288
#define T_FH_N 16384
#define T_GH_R 20480
#define T_GH_I 24576
#define T_GH_N 28672
#define T_GW_R 32768
#define T_GW_I 36864
#define T_TOTAL 40960

// ---------------------------------------------------------------------------
// Async global->LDS path (CDNA5 GLOBAL_LOAD_ASYNC_TO_LDS_B128 + ASYNCcnt).
// Builtin prototype (from hipcc diagnostic): (v4i AS1* src, v4i AS3* dst,
// imm offset, imm cpol).  Guarded so the build stays green if absent.
// ---------------------------------------------------------------------------
#if __has_builtin(__builtin_amdgcn_global_load_async_to_lds_b128) && \
    __has_builtin(__builtin_amdgcn_s_wait_asynccnt)
#define USE_ASYNC 1
#else
#define USE_ASYNC 0
#endif

#if USE_ASYNC
typedef __attribute__((address_space(1))) v4i gv4i_t;
typedef __attribute__((address_space(3))) v4i lv4i_t;
__device__ __forceinline__ void async_b128(const float* g, float* l) {
  __builtin_amdgcn_global_load_async_to_lds_b128((gv4i_t*)g, (lv4i_t*)l, 0, 0);
}
#endif

// ---------------------------------------------------------------------------
// WMMA helpers (fragment layouts per CDNA5 ISA 7.12.2)
// ---------------------------------------------------------------------------
__device__ __forceinline__ v8f wmma4(v2f a, v2f b, v8f c) {
  // D = A(16x4,f32) * B(4x16,f32) + C(16x16,f32)
  return __builtin_amdgcn_wmma_f32_16x16x4_f32(false, a, false, b, (short)0, c,
                                               false, false);
}
__device__ __forceinline__ v8f zero8() {
  v8f z = {0.f, 0.f, 0.f, 0.f, 0.f, 0.f, 0.f, 0.f};
  return z;
}
// A fragment: row = M index (0..15 within tile), kk = k0 + 2*half
__device__ __forceinline__ v2f ldA(const float* __restrict__ t, int row, int ld,
                                   int kk) {
  const float* p = t + row * ld + kk;
  v2f a; a.x = p[0]; a.y = p[1]; return a;
}
// B fragment: element (kk, col) and (kk+1, col), row stride ldb
__device__ __forceinline__ v2f ldB(const float* __restrict__ bm, int kk,
                                   size_t ldb, int col) {
  const float* p = bm + (size_t)kk * ldb + col;
  v2f b; b.x = p[0]; b.y = p[ldb]; return b;
}
__device__ __forceinline__ float gelu_tanh(float v) {
  const float c = 0.7978845608028654f;
  float t = tanhf(c * (v + 0.044715f * v * v * v));
  return 0.5f * v * (1.0f + t);
}

// ---------------------------------------------------------------------------
// Twiddle tables.  All regions are 4096 entries; angles use exact (p*q) mod 128.
// s = 1/sqrt(128) per 1D ortho stage.
// ---------------------------------------------------------------------------
__global__ __launch_bounds__(256) void k_init_tables(float* __restrict__ tab) {
  int i = blockIdx.x * blockDim.x + threadIdx.x;
  if (i >= 4096) return;
  const float s = 0.08838834764831845f;          // 1/sqrt(128)
  const float w = 0.049087385212340517f;         // 2*pi/128
  {   // forward tables, i = kw*128 + m (32x128 row-major, ld=128)
    int kw = i >> 7, m = i & 127;
    float th = w * (float)((kw * m) & 127);
    float sn, cs; __sincosf(th, &sn, &cs);
    tab[T_FW_R + i] =  cs * s;     // e^{-i th}: real
    tab[T_FW_I + i] = -sn * s;     // e^{-i th}: imag
    tab[T_FH_R + i] =  cs * s;
    tab[T_FH_I + i] = -sn * s;
    tab[T_FH_N + i] =  sn * s;     // negated imag (WMMA cannot negate B)
  }
  {   // inverse tables, i = n*32 + k (128x32 row-major, ld=32)
    int n = i >> 5, k = i & 31;
    float th = w * (float)((n * k) & 127);
    float sn, cs; __sincosf(th, &sn, &cs);
    tab[T_GH_R + i] =  cs * s;     // e^{+i th}
    tab[T_GH_I + i] =  sn * s;
    tab[T_GH_N + i] = -sn * s;
    float ck = (k == 0) ? 1.f : 2.f;   // Hermitian fold for irfft
    tab[T_GW_R + i] =  ck * cs * s;
    tab[T_GW_I + i] = -ck * sn * s;
  }
}

// ---------------------------------------------------------------------------
// Stage 1: truncated rDFT along w.  grid=(12,1024) block=128 (4 waves).
// Wave: one 16-col c tile, full M=32 (kw), r+i.  K=128.
// ---------------------------------------------------------------------------
__global__ __launch_bounds__(128) void k_stage1(const float* __restrict__ x,
                                                const float* __restrict__ tab,
                                                float* __restrict__ s1r,
                                                float* __restrict__ s1i) {
  const int wave = threadIdx.x >> 5, lane = threadIdx.x & 31;
  const int half = lane >> 4, mr = lane & 15;
  const int n0 = (blockIdx.x * 4 + wave) * 16;   // c tile
  const int bh = blockIdx.y;                     // b*128 + h
  const float* X  = x + (size_t)bh * 128 * C_;   // [w=128][c=768]
  const float* Fr = tab + T_FW_R;
  const float* Fi = tab + T_FW_I;
  v8f ar0 = zero8(), ar1 = zero8(), ai0 = zero8(), ai1 = zero8();
  for (int k0 = 0; k0 < 128; k0 += 4) {
    const int kk = k0 + 2 * half;
    const float* bp = X + (size_t)kk * C_ + n0 + mr;
    __builtin_prefetch(bp + 4 * C_, 0, 1);
    v2f b; b.x = bp[0]; b.y = bp[C_];
    v2f fr0 = ldA(Fr, mr,      128, kk);
    v2f fr1 = ldA(Fr, mr + 16, 128, kk);
    v2f fi0 = ldA(Fi, mr,      128, kk);
    v2f fi1 = ldA(Fi, mr + 16, 128, kk);
    ar0 = wmma4(fr0, b, ar0);
    ar1 = wmma4(fr1, b, ar1);
    ai0 = wmma4(fi0, b, ai0);
    ai1 = wmma4(fi1, b, ai1);
  }
  float* outr = s1r + (size_t)bh * KM_ * C_ + n0 + mr;
  float* outi = s1i + (size_t)bh * KM_ * C_ + n0 + mr;
#pragma unroll
  for (int r = 0; r < 8; ++r) {
    int row = r + 8 * half;
    outr[(size_t)row * C_]        = ar0[r];
    outr[(size_t)(row + 16) * C_] = ar1[r];
    outi[(size_t)row * C_]        = ai0[r];
    outi[(size_t)(row + 16) * C_] = ai1[r];
  }
}

// ---------------------------------------------------------------------------
// Stage 2: complex DFT along h.  grid=(384,8) block=128.
// Per b: S2 = Fh (x) S1, M=32(kh), K=128(h), N=24576(kw*c).
// B operand (S1 r/i) streamed via async global->LDS, double buffered:
// chunk = 8 K-rows x 16 cols x {r,i} = 1 KB/wave; issue chunk c+1, wait
// asynccnt<=2 (async loads complete in order), consume chunk c from LDS.
// ---------------------------------------------------------------------------
__global__ __launch_bounds__(128) void k_stage2(const float* __restrict__ s1r,
                                                const float* __restrict__ s1i,
                                                const float* __restrict__ tab,
                                                float* __restrict__ s2r,
                                                float* __restrict__ s2i) {
  const int wave = threadIdx.x >> 5, lane = threadIdx.x & 31;
  const int half = lane >> 4, mr = lane & 15;
  const int n0 = (blockIdx.x * 4 + wave) * 16;   // 0..24560
  const int b  = blockIdx.y;
  const float* Br = s1r + (size_t)b * 128 * N2_;  // [h][kw*c], ld=N2_
  const float* Bi = s1i + (size_t)b * 128 * N2_;
  const float* Hr = tab + T_FH_R;
  const float* Hi = tab + T_FH_I;
  const float* Hn = tab + T_FH_N;
  v8f rr0 = zero8(), rr1 = zero8(), ii0 = zero8(), ii1 = zero8();

#if USE_ASYNC
  __shared__ float bstage[4][2][2][8][16];   // [wave][buf][ri][row][col] 8 KB
  const int arow = lane >> 2;                // 0..7   (K row within chunk)
  const int ac4  = (lane & 3) * 4;           // 0,4,8,12 (col within tile)
  // issue one chunk: two b128 async ops (r and i planes)
  auto issue = [&](int k0, int buf) {
    const float* gr = Br + (size_t)(k0 + arow) * N2_ + n0 + ac4;
    const float* gi = Bi + (size_t)(k0 + arow) * N2_ + n0 + ac4;
    async_b128(gr, &bstage[wave][buf][0][arow][ac4]);
    async_b128(gi, &bstage[wave][buf][1][arow][ac4]);
  };
  issue(0, 0);
  int buf = 0;
  for (int c = 0; c < 16; ++c) {
    const int k0 = c * 8;
    if (c < 15) {
      issue(k0 + 8, buf ^ 1);
      __builtin_amdgcn_s_wait_asynccnt(2);   // chunk c resident
    } else {
      __builtin_amdgcn_s_wait_asynccnt(0);
    }
#pragma unroll
    for (int sub = 0; sub < 2; ++sub) {
      const int rloc = sub * 4 + 2 * half;   // row within chunk
      const int kk   = k0 + rloc;
      v2f br; br.x = bstage[wave][buf][0][rloc][mr];
              br.y = bstage[wave][buf][0][rloc + 1][mr];
      v2f bi; bi.x = bstage[wave][buf][1][rloc][mr];
              bi.y = bstage[wave][buf][1][rloc + 1][mr];
      v2f hr0 = ldA(Hr, mr, 128, kk), hr1 = ldA(Hr, mr + 16, 128, kk);
      v2f hi0 = ldA(Hi, mr, 128, kk), hi1 = ldA(Hi, mr + 16, 128, kk);
      v2f hn0 = ldA(Hn, mr, 128, kk), hn1 = ldA(Hn, mr + 16, 128, kk);
      rr0 = wmma4(hr0, br, rr0);  rr0 = wmma4(hn0, bi, rr0);
      rr1 = wmma4(hr1, br, rr1);  rr1 = wmma4(hn1, bi, rr1);
      ii0 = wmma4(hr0, bi, ii0);  ii0 = wmma4(hi0, br, ii0);
      ii1 = wmma4(hr1, bi, ii1);  ii1 = wmma4(hi1, br, ii1);
    }
    buf ^= 1;
  }
#else
  for (int k0 = 0; k0 < 128; k0 += 4) {
    const int kk = k0 + 2 * half;
    const float* pr = Br + (size_t)kk * N2_ + n0 + mr;
    const float* pi = Bi + (size_t)kk * N2_ + n0 + mr;
    __builtin_prefetch(pr + 4 * (size_t)N2_, 0, 1);
    __builtin_prefetch(pi + 4 * (size_t)N2_, 0, 1);
    v2f br; br.x = pr[0]; br.y = pr[N2_];
    v2f bi; bi.x = pi[0]; bi.y = pi[N2_];
    v2f hr0 = ldA(Hr, mr, 128, kk), hr1 = ldA(Hr, mr + 16, 128, kk);
    v2f hi0 = ldA(Hi, mr, 128, kk), hi1 = ldA(Hi, mr + 16, 128, kk);
    v2f hn0 = ldA(Hn, mr, 128, kk), hn1 = ldA(Hn, mr + 16, 128, kk);
    rr0 = wmma4(hr0, br, rr0);  rr0 = wmma4(hn0, bi, rr0);
    rr1 = wmma4(hr1, br, rr1);  rr1 = wmma4(hn1, bi, rr1);
    ii0 = wmma4(hr0, bi, ii0);  ii0 = wmma4(hi0, br, ii0);
    ii1 = wmma4(hr1, bi, ii1);  ii1 = wmma4(hi1, br, ii1);
  }
#endif

  float* outr = s2r + (size_t)b * KM_ * N2_ + n0 + mr;
  float* outi = s2i + (size_t)b * KM_ * N2_ + n0 + mr;
#pragma unroll
  for (int r = 0; r < 8; ++r) {
    int row = r + 8 * half;
    outr[(size_t)row * N2_]        = rr0[r];
    outr[(size_t)(row + 16) * N2_] = rr1[r];
    outi[(size_t)row * N2_]        = ii0[r];
    outi[(size_t)(row + 16) * N2_] = ii1[r];
  }
}

// ---------------------------------------------------------------------------
// Stage 3: block-diagonal complex MLP (two layers, GELU between).
// grid=(128, 8 blocks), block=128.  Wave: 16 spectral points, one channel
// block.  Layer-1 output staged in LDS (per-wave region, 48 KB total).
// ---------------------------------------------------------------------------
__global__ __launch_bounds__(128) void k_stage3(const float* __restrict__ s2r,
                                                const float* __restrict__ s2i,
                                                const float* __restrict__ w1,
                                                const float* __restrict__ b1,
                                                const float* __restrict__ w2,
                                                const float* __restrict__ b2,
                                                float* __restrict__ s3r,
                                                float* __restrict__ s3i) {
  __shared__ float lds[4 * 16 * BS_ * 2];   // 12288 floats = 48 KB
  const int wave = threadIdx.x >> 5, lane = threadIdx.x & 31;
  const int half = lane >> 4, mr = lane & 15;
  const int blk = blockIdx.y;
  const int p0  = (blockIdx.x * 4 + wave) * 16;     // spectral point tile
  const size_t cb = (size_t)blk * BS_;
  const float* W1r = w1 + (size_t)blk * BS_ * BS_;
  const float* W1i = w1 + (size_t)(NB_BLK + blk) * BS_ * BS_;
  const float* W2r = w2 + (size_t)blk * BS_ * BS_;
  const float* W2i = w2 + (size_t)(NB_BLK + blk) * BS_ * BS_;
  const float* B1r = b1 + blk * BS_;
  const float* B1i = b1 + (NB_BLK + blk) * BS_;
  const float* B2r = b2 + blk * BS_;
  const float* B2i = b2 + (NB_BLK + blk) * BS_;
  float* o1r = &lds[wave * 16 * BS_ * 2];
  float* o1i = o1r + 16 * BS_;

  // ---- layer 1: o1 = gelu(Xr*W1r - Xi*W1i + b1r , Xi*W1r + Xr*W1i + b1i)
  for (int nt = 0; nt < 6; ++nt) {
    const int n0 = nt * 16;
    v8f accr = zero8(), acci = zero8();
    for (int k0 = 0; k0 < BS_; k0 += 4) {
      const int kk = k0 + 2 * half;
      const float* pr = s2r + (size_t)(p0 + mr) * C_ + cb + kk;
      const float* pi = s2i + (size_t)(p0 + mr) * C_ + cb + kk;
      v2f axr; axr.x = pr[0]; axr.y = pr[1];
      v2f axi; axi.x = pi[0]; axi.y = pi[1];
      v2f axn; axn.x = -axi.x; axn.y = -axi.y;   // negate A for -Xi*W1i
      v2f bwr = ldB(W1r, kk, BS_, n0 + mr);
      v2f bwi = ldB(W1i, kk, BS_, n0 + mr);
      accr = wmma4(axr, bwr, accr);  accr = wmma4(axn, bwi, accr);
      acci = wmma4(axi, bwr, acci);  acci = wmma4(axr, bwi, acci);
    }
    const float br = B1r[n0 + mr], bi = B1i[n0 + mr];
#pragma unroll
    for (int r = 0; r < 8; ++r) {
      int row = r + 8 * half;
      o1r[row * BS_ + n0 + mr] = gelu_tanh(accr[r] + br);
      o1i[row * BS_ + n0 + mr] = gelu_tanh(acci[r] + bi);
    }
  }

  // ---- layer 2: o2 = o1r*W2r - o1i*W2i + b2r , o1i*W2r + o1r*W2i + b2i
  for (int nt = 0; nt < 6; ++nt) {
    const int n0 = nt * 16;
    v8f accr = zero8(), acci = zero8();
    for (int k0 = 0; k0 < BS_; k0 += 4) {
      const int kk = k0 + 2 * half;
      v2f a1r; a1r.x = o1r[mr * BS_ + kk]; a1r.y = o1r[mr * BS_ + kk + 1];
      v2f a1i; a1i.x = o1i[mr * BS_ + kk]; a1i.y = o1i[mr * BS_ + kk + 1];
      v2f a1n; a1n.x = -a1i.x; a1n.y = -a1i.y;
      v2f bwr = ldB(W2r, kk, BS_, n0 + mr);
      v2f bwi = ldB(W2i, kk, BS_, n0 + mr);
      accr = wmma4(a1r, bwr, accr);  accr = wmma4(a1n, bwi, accr);
      acci = wmma4(a1i, bwr, acci);  acci = wmma4(a1r, bwi, acci);
    }
    const float br = B2r[n0 + mr], bi = B2i[n0 + mr];
#pragma unroll
    for (int r = 0; r < 8; ++r) {
      int row = r + 8 * half;
      size_t o = (size_t)(p0 + row) * C_ + cb + n0 + mr;
      s3r[o] = accr[r] + br;
      s3i[o] = acci[r] + bi;
    }
  }
}

// ---------------------------------------------------------------------------
// Stage 4: inverse DFT along h from 32 modes.  grid=(384,8,2) block=128.
// Per b: S4 = Gh (x) S3, M=128(h) in 2 groups of 4 m-tiles, K=32, N=24576.
// ---------------------------------------------------------------------------
__global__ __launch_bounds__(128) void k_stage4(const float* __restrict__ s3r,
                                                const float* __restrict__ s3i,
                                                const float* __restrict__ tab,
                                                float* __restrict__ s4r,
                                                float* __restrict__ s4i) {
  const int wave = threadIdx.x >> 5, lane = threadIdx.x & 31;
  const int half = lane >> 4, mr = lane & 15;
  const int n0 = (blockIdx.x * 4 + wave) * 16;
  const int b  = blockIdx.y;
  const int mg = blockIdx.z;                     // h group: 0..63 / 64..127
  const float* Br = s3r + (size_t)b * KM_ * N2_;  // [kh][kw*c]
  const float* Bi = s3i + (size_t)b * KM_ * N2_;
  const float* Gr = tab + T_GH_R;
  const float* Gi = tab + T_GH_I;
  const float* Gn = tab + T_GH_N;
  v8f accr[4], acci[4];
#pragma unroll
  for (int mt = 0; mt < 4; ++mt) { accr[mt] = zero8(); acci[mt] = zero8(); }
  for (int k0 = 0; k0 < KM_; k0 += 4) {
    const int kk = k0 + 2 * half;
    const float* pr = Br + (size_t)kk * N2_ + n0 + mr;
    const float* pi = Bi + (size_t)kk * N2_ + n0 + mr;
    v2f br; br.x = pr[0]; br.y = pr[N2_];
    v2f bi; bi.x = pi[0]; bi.y = pi[N2_];
#pragma unroll
    for (int mt = 0; mt < 4; ++mt) {
      const int mrow = mg * 64 + mt * 16 + mr;
      v2f gr = ldA(Gr, mrow, KM_, kk);
      v2f gi = ldA(Gi, mrow, KM_, kk);
      v2f gn = ldA(Gn, mrow, KM_, kk);
      accr[mt] = wmma4(gr, br, accr[mt]);  accr[mt] = wmma4(gn, bi, accr[mt]);
      acci[mt] = wmma4(gr, bi, acci[mt]);  acci[mt] = wmma4(gi, br, acci[mt]);
    }
  }
#pragma unroll
  for (int mt = 0; mt < 4; ++mt) {
#pragma unroll
    for (int r = 0; r < 8; ++r) {
      int nh = mg * 64 + mt * 16 + r + 8 * half;
      size_t o = (size_t)b * 128 * N2_ + (size_t)nh * N2_ + n0 + mr;
      s4r[o] = accr[mt][r];
      s4i[o] = acci[mt][r];
    }
  }
}

// ---------------------------------------------------------------------------
// Stage 5: inverse real DFT along w (Hermitian-folded) + residual.
// grid=(12,1024,2) block=128.  Per (b,h): out[128x768] = Gwr*S4r + Gwi*S4i + x
// ---------------------------------------------------------------------------
__global__ __launch_bounds__(128) void k_stage5(const float* __restrict__ s4r,
                                                const float* __restrict__ s4i,
                                                const float* __restrict__ tab,
                                                const float* __restrict__ x,
                                                float* __restrict__ out) {
  const int wave = threadIdx.x >> 5, lane = threadIdx.x & 31;
  const int half = lane >> 4, mr = lane & 15;
  const int n0 = (blockIdx.x * 4 + wave) * 16;   // c tile
  const int bh = blockIdx.y;                     // b*128 + h
  const int mg = blockIdx.z;                     // w group
  const float* Br = s4r + (size_t)bh * N2_;       // [kw][c], ld=C_
  const float* Bi = s4i + (size_t)bh * N2_;
  const float* Gr = tab + T_GW_R;
  const float* Gi = tab + T_GW_I;
  v8f acc[4];
#pragma unroll
  for (int mt = 0; mt < 4; ++mt) acc[mt] = zero8();
  for (int k0 = 0; k0 < KM_; k0 += 4) {
    const int kk = k0 + 2 * half;
    const float* pr = Br + (size_t)kk * C_ + n0 + mr;
    const float* pi = Bi + (size_t)kk * C_ + n0 + mr;
    v2f br; br.x = pr[0]; br.y = pr[C_];
    v2f bi; bi.x = pi[0]; bi.y = pi[C_];
#pragma unroll
    for (int mt = 0; mt < 4; ++mt) {
      const int mrow = mg * 64 + mt * 16 + mr;
      v2f gr = ldA(Gr, mrow, KM_, kk);
      v2f gi = ldA(Gi, mrow, KM_, kk);
      acc[mt] = wmma4(gr, br, acc[mt]);
      acc[mt] = wmma4(gi, bi, acc[mt]);
    }
  }
#pragma unroll
  for (int mt = 0; mt < 4; ++mt) {
#pragma unroll
    for (int r = 0; r < 8; ++r) {
      int wrow = mg * 64 + mt * 16 + r + 8 * half;
      size_t o = ((size_t)bh * 128 + wrow) * C_ + n0 + mr;
      out[o] = acc[mt][r] + x[o];
    }
  }
}

// ---------------------------------------------------------------------------
extern "C" void kernel_launch(void* const* d_in, const int* in_sizes, int n_in,
                              void* d_out, int out_size, void* d_ws,
                              size_t ws_size, hipStream_t stream) {
  const float* x  = (const float*)d_in[0];
  const float* w1 = (const float*)d_in[1];
  const float* b1 = (const float*)d_in[2];
  const float* w2 = (const float*)d_in[3];
  const float* b2 = (const float*)d_in[4];
  float* out = (float*)d_out;
  float* ws  = (float*)d_ws;

  const size_t S2N = (size_t)8 * KM_ * N2_;        // 6,291,456 floats per buf
  const size_t S1N = (size_t)8 * 128 * KM_ * C_;   // 25,165,824 floats per buf
  const size_t need = (65536 + 4 * S2N + 2 * S1N) * sizeof(float); // ~302 MB
  if (ws_size < need) return;

  float* tab = ws;
  float* s2r = ws + 65536;
  float* s2i = s2r + S2N;
  float* s3r = s2i + S2N;
  float* s3i = s3r + S2N;
  float* s1r = s3i + S2N;
  float* s1i = s1r + S1N;
  float* s4r = s1r;   // S1 is dead after stage2 -> alias for S4
  float* s4i = s1i;

  k_init_tables<<<dim3(16), dim3(256), 0, stream>>>(tab);
  k_stage1<<<dim3(12, 1024), dim3(128), 0, stream>>>(x, tab, s1r, s1i);
  k_stage2<<<dim3(384, 8), dim3(128), 0, stream>>>(s1r, s1i, tab, s2r, s2i);
  k_stage3<<<dim3(128, 8), dim3(128), 0, stream>>>(s2r, s2i, w1, b1, w2, b2,
                                                   s3r, s3i);
  k_stage4<<<dim3(384, 8, 2), dim3(128), 0, stream>>>(s3r, s3i, tab, s4r, s4i);
  k_stage5<<<dim3(12, 1024, 2), dim3(128), 0, stream>>>(s4r, s4i, tab, x, out);
}